// DeltaNet_31877247271436
// MI455X (gfx1250) — compile-verified
//
#include <hip/hip_runtime.h>

typedef __attribute__((ext_vector_type(2))) float v2f;
typedef __attribute__((ext_vector_type(8))) float v8f;

// D = A(16x4) x B(4x16) + C, fp32 WMMA (CDNA5)
#define WMMA_F32(a, b, c) \
  __builtin_amdgcn_wmma_f32_16x16x4_f32(false, (a), false, (b), (short)0, (c), false, false)

// Async global->LDS copy (CDNA5): per-lane LDS scatter, tracked by ASYNCcnt.
__device__ __forceinline__ void async_b128(const float* gsrc, const float* ldst) {
  uint32_t lo = (uint32_t)(uintptr_t)ldst;  // low 32 bits of generic LDS addr = LDS offset
  asm volatile("global_load_async_to_lds_b128 %0, %1, off"
               :: "v"(lo), "v"(gsrc) : "memory");
}
#define ASYNC_WAIT(n) asm volatile("s_wait_asynccnt " #n ::: "memory")

constexpr int B_  = 4;
constexpr int H_  = 8;
constexpr int L_  = 8192;
constexpr int D_  = 64;
constexpr int C_  = 32;
constexpr int NCH = L_ / C_;  // 256 chunks per (b,h)
constexpr int BH  = B_ * H_;  // 32

// Padded strides: 68 keeps B128 LDS destinations 16B-aligned (272*r + 4c) and
// fragment loads bank-conflict-free (banks (4m+kk) mod 64 distinct). 36 ditto
// for 32x32 tiles.
constexpr int P64 = 68;
constexpr int P32 = 36;

// ---------------------------------------------------------------------------
// Phase 1: per-chunk (fully parallel, 8192 blocks):
//   kn = l2norm(k); kb = kn*beta; A = I + tril(kb @ kn^T, -1); T = A^{-1}
// Store T (32x32) to workspace.
// ---------------------------------------------------------------------------
__global__ void __launch_bounds__(128)
dn_phase1(const float* __restrict__ kg, const float* __restrict__ betag,
          float* __restrict__ Tg) {
  const int cg   = blockIdx.x;             // bh*NCH + chunk
  const int tid  = threadIdx.x;
  const int lane = tid & 31, wave = tid >> 5;
  const int half = lane >> 4, ln = lane & 15;

  __shared__ float sK[32 * P64];
  __shared__ float sKB[32 * P64];
  __shared__ float sA[32 * P32];
  __shared__ float sT[32 * P32];
  __shared__ float sScale[32];
  __shared__ float sBeta[32];

  const float* kp = kg + (size_t)cg * (C_ * D_);
  for (int idx = tid; idx < C_ * D_; idx += 128)
    sK[(idx >> 6) * P64 + (idx & 63)] = kp[idx];
  __syncthreads();

  if (tid < 32) {
    float s = 0.f;
    for (int c = 0; c < 64; ++c) { float x = sK[tid * P64 + c]; s += x * x; }
    sScale[tid] = rsqrtf(s + 1e-6f);
    sBeta[tid]  = betag[(size_t)cg * C_ + tid];
  }
  __syncthreads();

  for (int idx = tid; idx < C_ * D_; idx += 128) {
    int r = idx >> 6, c = idx & 63;
    float kn = sK[r * P64 + c] * sScale[r];
    sK[r * P64 + c]  = kn;
    sKB[r * P64 + c] = kn * sBeta[r];
  }
  __syncthreads();

  // A = I + tril(kb @ kn^T, -1): 4 tiles of 16x16, one per wave, K=64
  {
    const int row0 = (wave >> 1) * 16, col0 = (wave & 1) * 16;
    v8f acc = {};
    for (int k0 = 0; k0 < 64; k0 += 4) {
      const int kk = k0 + 2 * half;
      v2f a, b;
      a.x = sKB[(row0 + ln) * P64 + kk]; a.y = sKB[(row0 + ln) * P64 + kk + 1];
      b.x = sK[(col0 + ln) * P64 + kk]; b.y = sK[(col0 + ln) * P64 + kk + 1];
      acc = WMMA_F32(a, b, acc);
    }
    for (int j = 0; j < 8; ++j) {
      int mg = row0 + j + 8 * half, ng = col0 + ln;
      sA[mg * P32 + ng] = (mg > ng) ? acc[j] : (mg == ng ? 1.f : 0.f);
    }
  }
  __syncthreads();

  // Unit-lower-triangular inverse, forward substitution: one column per thread
  if (tid < 32) {
    const int j = tid;
    for (int i = 0; i < j; ++i) sT[i * P32 + j] = 0.f;
    sT[j * P32 + j] = 1.f;
    for (int i = j + 1; i < 32; ++i) {
      float s = 0.f;
      for (int m = j; m < i; ++m) s += sA[i * P32 + m] * sT[m * P32 + j];
      sT[i * P32 + j] = -s;
    }
  }
  __syncthreads();

  float* tp = Tg + (size_t)cg * (C_ * C_);
  for (int idx = tid; idx < C_ * C_; idx += 128)
    tp[idx] = sT[(idx >> 5) * P32 + (idx & 31)];
}

// ---------------------------------------------------------------------------
// Phase 2: per-(b,h) sequential scan over 256 chunks (32 blocks x 8 waves).
// S (64x64) lives in LDS. Next chunk's q/k/v/T are prefetched into a second
// LDS buffer with async global->LDS loads (ASYNCcnt) while the current chunk
// computes. Per chunk:
//   qn,kn = l2norm; vb = v*beta; kb = kn*beta
//   u = T@vb; w = T@kb; attn = tril(qn@kn^T)
//   u_adj = u - w@S; o = qn@S + attn@u_adj; S += kn^T @ u_adj
// ---------------------------------------------------------------------------
__global__ void __launch_bounds__(256)
dn_phase2(const float* __restrict__ qg, const float* __restrict__ kg,
          const float* __restrict__ vg, const float* __restrict__ betag,
          const float* __restrict__ Tg, float* __restrict__ og) {
  const int bh   = blockIdx.x;
  const int tid  = threadIdx.x;
  const int lane = tid & 31, wave = tid >> 5;
  const int half = lane >> 4, ln = lane & 15;

  __shared__ float sS[64 * P64];
  __shared__ float sQ[2][32 * P64], sK[2][32 * P64], sV[2][32 * P64];
  __shared__ float sKB[32 * P64], sU[32 * P64], sW[32 * P64], sUA[32 * P64];
  __shared__ float sT[2][32 * P32], sAt[32 * P32];
  __shared__ float sNq[32], sNk[32], sBeta[32];

  const size_t base = (size_t)bh * L_ * D_;

  // Issue one chunk's async loads: 7 x B128 per thread (q,k,v: 2 each; T: 1).
  auto issue = [&](int chn, int b) {
    const float* qp = qg + base + (size_t)chn * (C_ * D_);
    const float* kp = kg + base + (size_t)chn * (C_ * D_);
    const float* vp = vg + base + (size_t)chn * (C_ * D_);
    const float* tp = Tg + ((size_t)bh * NCH + chn) * (C_ * C_);
#pragma unroll
    for (int rep = 0; rep < 2; ++rep) {
      int g = tid + rep * 256;             // float4 index 0..511
      int r = g >> 4, c = (g & 15) << 2;   // row 0..31, col 0..60
      int off = r * P64 + c;
      async_b128(qp + 4 * g, &sQ[b][off]);
      async_b128(kp + 4 * g, &sK[b][off]);
      async_b128(vp + 4 * g, &sV[b][off]);
    }
    {
      int r = tid >> 3, c = (tid & 7) << 2;
      async_b128(tp + 4 * tid, &sT[b][r * P32 + c]);
    }
  };

  for (int idx = tid; idx < 64 * P64; idx += 256) sS[idx] = 0.f;

  issue(0, 0);  // prefetch chunk 0

  for (int ch = 0; ch < NCH; ++ch) {
    const int buf = ch & 1;

    // Prefetch next chunk, then wait for current chunk's 7 transfers
    // (async loads retire in order, so ASYNCcnt<=7 => oldest 7 complete).
    if (ch + 1 < NCH) { issue(ch + 1, buf ^ 1); ASYNC_WAIT(7); }
    else              { ASYNC_WAIT(0); }
    __syncthreads();

    // Row norms: 8 threads per row + shfl_xor reduction.
    {
      int r = tid >> 3, s8 = tid & 7;
      const float* qr = &sQ[buf][r * P64 + s8 * 8];
      const float* kr = &sK[buf][r * P64 + s8 * 8];
      float sq = 0.f, sk = 0.f;
      for (int c = 0; c < 8; ++c) {
        float xq = qr[c]; sq += xq * xq;
        float xk = kr[c]; sk += xk * xk;
      }
      sq += __shfl_xor(sq, 1); sq += __shfl_xor(sq, 2); sq += __shfl_xor(sq, 4);
      sk += __shfl_xor(sk, 1); sk += __shfl_xor(sk, 2); sk += __shfl_xor(sk, 4);
      if (s8 == 0) { sNq[r] = rsqrtf(sq + 1e-6f); sNk[r] = rsqrtf(sk + 1e-6f); }
      if (tid < 32) sBeta[tid] = betag[(size_t)bh * L_ + ch * C_ + tid];
    }
    __syncthreads();

    for (int idx = tid; idx < C_ * D_; idx += 256) {
      int r = idx >> 6, c = idx & 63;
      int o = r * P64 + c;
      sQ[buf][o] *= sNq[r];
      float kn = sK[buf][o] * sNk[r];
      sK[buf][o] = kn;
      sKB[o]     = kn * sBeta[r];
      sV[buf][o] *= sBeta[r];
    }
    __syncthreads();

    // u = T @ vb ; w = T @ kb : 8 tiles of 32x64, one tile per wave, K=32
    {
      const int row0 = (wave >> 2) * 16, col0 = (wave & 3) * 16;
      v8f au = {}, aw = {};
      for (int k0 = 0; k0 < 32; k0 += 4) {
        const int kk = k0 + 2 * half;
        v2f a, bu, bw;
        a.x  = sT[buf][(row0 + ln) * P32 + kk];
        a.y  = sT[buf][(row0 + ln) * P32 + kk + 1];
        bu.x = sV[buf][kk * P64 + col0 + ln];
        bu.y = sV[buf][(kk + 1) * P64 + col0 + ln];
        bw.x = sKB[kk * P64 + col0 + ln];
        bw.y = sKB[(kk + 1) * P64 + col0 + ln];
        au = WMMA_F32(a, bu, au);
        aw = WMMA_F32(a, bw, aw);
      }
      for (int j = 0; j < 8; ++j) {
        int mg = row0 + j + 8 * half, ng = col0 + ln;
        sU[mg * P64 + ng] = au[j];
        sW[mg * P64 + ng] = aw[j];
      }
    }
    // attn = tril(qn @ kn^T) incl diag : 4 tiles, waves 0..3, K=64
    if (wave < 4) {
      const int row0 = (wave >> 1) * 16, col0 = (wave & 1) * 16;
      v8f acc = {};
      for (int k0 = 0; k0 < 64; k0 += 4) {
        const int kk = k0 + 2 * half;
        v2f a, b;
        a.x = sQ[buf][(row0 + ln) * P64 + kk]; a.y = sQ[buf][(row0 + ln) * P64 + kk + 1];
        b.x = sK[buf][(col0 + ln) * P64 + kk]; b.y = sK[buf][(col0 + ln) * P64 + kk + 1];
        acc = WMMA_F32(a, b, acc);
      }
      for (int j = 0; j < 8; ++j) {
        int mg = row0 + j + 8 * half, ng = col0 + ln;
        sAt[mg * P32 + ng] = (mg >= ng) ? acc[j] : 0.f;
      }
    }
    __syncthreads();

    // u_adj = u - w@S  (accumulate M = w@S - u, store -M); K=64
    {
      const int row0 = (wave >> 2) * 16, col0 = (wave & 3) * 16;
      v8f acc;
      for (int j = 0; j < 8; ++j)
        acc[j] = -sU[(row0 + j + 8 * half) * P64 + col0 + ln];
      for (int k0 = 0; k0 < 64; k0 += 4) {
        const int kk = k0 + 2 * half;
        v2f a, b;
        a.x = sW[(row0 + ln) * P64 + kk]; a.y = sW[(row0 + ln) * P64 + kk + 1];
        b.x = sS[kk * P64 + col0 + ln];   b.y = sS[(kk + 1) * P64 + col0 + ln];
        acc = WMMA_F32(a, b, acc);
      }
      for (int j = 0; j < 8; ++j)
        sUA[(row0 + j + 8 * half) * P64 + col0 + ln] = -acc[j];
    }
    __syncthreads();

    // o = qn@S + attn@u_adj -> global
    {
      const int row0 = (wave >> 2) * 16, col0 = (wave & 3) * 16;
      v8f acc = {};
      for (int k0 = 0; k0 < 64; k0 += 4) {           // qn @ S, K=64
        const int kk = k0 + 2 * half;
        v2f a, b;
        a.x = sQ[buf][(row0 + ln) * P64 + kk]; a.y = sQ[buf][(row0 + ln) * P64 + kk + 1];
        b.x = sS[kk * P64 + col0 + ln];        b.y = sS[(kk + 1) * P64 + col0 + ln];
        acc = WMMA_F32(a, b, acc);
      }
      for (int k0 = 0; k0 < 32; k0 += 4) {           // attn @ u_adj, K=32
        const int kk = k0 + 2 * half;
        v2f a, b;
        a.x = sAt[(row0 + ln) * P32 + kk]; a.y = sAt[(row0 + ln) * P32 + kk + 1];
        b.x = sUA[kk * P64 + col0 + ln];   b.y = sUA[(kk + 1) * P64 + col0 + ln];
        acc = WMMA_F32(a, b, acc);
      }
      float* op = og + base + (size_t)ch * (C_ * D_);
      for (int j = 0; j < 8; ++j)
        op[(row0 + j + 8 * half) * 64 + col0 + ln] = acc[j];
    }
    __syncthreads();  // all reads of S complete before update

    // S += kn^T @ u_adj : 16 tiles of 64x64, 2 per wave, K=32 (A transposed)
    for (int t = wave; t < 16; t += 8) {
      const int row0 = (t >> 2) * 16, col0 = (t & 3) * 16;
      v8f acc;
      for (int j = 0; j < 8; ++j)
        acc[j] = sS[(row0 + j + 8 * half) * P64 + col0 + ln];
      for (int k0 = 0; k0 < 32; k0 += 4) {
        const int kk = k0 + 2 * half;
        v2f a, b;
        a.x = sK[buf][kk * P64 + row0 + ln]; a.y = sK[buf][(kk + 1) * P64 + row0 + ln];
        b.x = sUA[kk * P64 + col0 + ln];     b.y = sUA[(kk + 1) * P64 + col0 + ln];
        acc = WMMA_F32(a, b, acc);
      }
      for (int j = 0; j < 8; ++j)
        sS[(row0 + j + 8 * half) * P64 + col0 + ln] = acc[j];
    }
    __syncthreads();  // S writes + buffer reads done before next iteration
  }
}

extern "C" void kernel_launch(void* const* d_in, const int* in_sizes, int n_in,
                              void* d_out, int out_size, void* d_ws, size_t ws_size,
                              hipStream_t stream) {
  (void)in_sizes; (void)n_in; (void)out_size; (void)ws_size;
  const float* q    = (const float*)d_in[0];
  const float* k    = (const float*)d_in[1];
  const float* v    = (const float*)d_in[2];
  const float* beta = (const float*)d_in[3];
  float* Tws = (float*)d_ws;  // needs BH*NCH*C*C*4 = 33.5 MB

  dn_phase1<<<BH * NCH, 128, 0, stream>>>(k, beta, Tws);
  dn_phase2<<<BH, 256, 0, stream>>>(q, k, v, beta, Tws, (float*)d_out);
}